// XPBDProjector_43860206026934
// MI455X (gfx1250) — compile-verified
//
#include <hip/hip_runtime.h>

#define N_ITERS   6
#define MAX_CORR  0.2f
#define EDGE_TILE 256

typedef unsigned int u32;
typedef __attribute__((ext_vector_type(4))) unsigned int u32x4;
typedef __attribute__((ext_vector_type(4))) int          i32x4;
typedef __attribute__((ext_vector_type(8))) int          i32x8;

#if defined(__HIP_DEVICE_COMPILE__) && defined(__gfx1250__) && \
    __has_builtin(__builtin_amdgcn_tensor_load_to_lds) &&      \
    __has_builtin(__builtin_amdgcn_s_wait_tensorcnt)
#define USE_TDM 1
#else
#define USE_TDM 0
#endif

// ---------------------------------------------------------------------------
// init: d_out = x_in ; delta = 0      (vectorized float4, B*N*3 % 4 == 0)
// ---------------------------------------------------------------------------
__global__ void xpbd_init_kernel(const float4* __restrict__ xin,
                                 float4* __restrict__ xout,
                                 float4* __restrict__ delta,
                                 int n4) {
    int i = blockIdx.x * blockDim.x + threadIdx.x;
    if (i < n4) {
        xout[i]  = xin[i];
        delta[i] = make_float4(0.f, 0.f, 0.f, 0.f);
    }
}

// ---------------------------------------------------------------------------
// edge phase: one thread per edge, loop over batches.
// The block's (2 x EDGE_TILE) tile of edge_index (src row / dst row, row
// stride = E) is DMA'd into LDS with one TENSOR_LOAD_TO_LDS per block
// (tensor_dim0 = E - e0 gives the ragged tail OOB->zero for free).
// Scatter uses non-returning global_atomic_add_f32 into delta.
// ---------------------------------------------------------------------------
__global__ void __launch_bounds__(EDGE_TILE)
xpbd_edge_kernel(const float* __restrict__ x,      // [B*N*3]
                 const int*   __restrict__ eidx,   // [2*E]: row0=src, row1=dst
                 const float* __restrict__ L0_a,   // [E]
                 const float* __restrict__ invm,   // [N]
                 float*       __restrict__ delta,  // [B*N*3]
                 int N, int E, int Bcnt) {
    __shared__ int lds_edge[2 * EDGE_TILE];   // [0..255]=src tile, [256..511]=dst tile

    const int e0 = blockIdx.x * EDGE_TILE;
    const int e  = e0 + threadIdx.x;

#if USE_TDM
    if (threadIdx.x < 32) {   // wave 0 (wave32) issues the tensor DMA for the block
        const unsigned long long ga =
            (unsigned long long)(const void*)(eidx + e0);   // tile start (row 0)
        const u32 lds_off = (u32)(unsigned long long)
            (__attribute__((address_space(3))) const int*)lds_edge;
        const u32 tdim0 = (u32)(E - e0);      // remaining cols -> tail reads zero

        // D# group 0: count=1 | lds_addr | global_addr[56:0] | type=2
        const u32x4 g0 = {
            1u,
            lds_off,
            (u32)(ga & 0xFFFFFFFFu),
            (u32)((ga >> 32) & 0x1FFFFFFu) | (2u << 30)
        };
        // D# group 1: data_size=4B; tensor = tdim0 x 2, tile = 256 x 2,
        //             tensor_dim0_stride = E (row pitch of edge_index)
        const i32x8 g1 = {
            (int)0x00020000u,                                   // data_size=2 (4B)
            (int)((tdim0 & 0xFFFFu) << 16),                     // tensor_dim0 lo
            (int)(((tdim0 >> 16) & 0xFFFFu) | (2u << 16)),      // dim0 hi | tensor_dim1=2
            (int)((u32)EDGE_TILE << 16),                        // tile_dim0=256
            (int)2,                                             // tile_dim1=2
            (int)(u32)E,                                        // tensor_dim0_stride lo
            0, 0                                                // stride hi / dim1_stride
        };
        const i32x4 z4 = 0;
#if defined(__clang_major__) && (__clang_major__ >= 23)
        const i32x8 z8 = 0;
        __builtin_amdgcn_tensor_load_to_lds(g0, g1, z4, z4, z8, 0);
#else
        __builtin_amdgcn_tensor_load_to_lds(g0, g1, z4, z4, 0);
#endif
        __builtin_amdgcn_s_wait_tensorcnt(0);
    }
    __syncthreads();
#else
    lds_edge[threadIdx.x]             = (e < E) ? eidx[e]     : 0;
    lds_edge[EDGE_TILE + threadIdx.x] = (e < E) ? eidx[E + e] : 0;
    __syncthreads();
#endif

    if (e >= E) return;

    const int   s  = lds_edge[threadIdx.x];
    const int   d  = lds_edge[EDGE_TILE + threadIdx.x];
    const float L0 = L0_a[e];
    const float wi = invm[s];
    const float wj = invm[d];
    // alpha = COMPLIANCE/ITERS = 0 ; fold d_lambda's sign into the factor
    const float neg_inv_denom = -1.0f / (wi + wj + 1e-9f);

    const long long so  = 3LL * s;
    const long long dof = 3LL * d;
    const long long N3  = 3LL * N;

    // Establish MLP: prefetch every gather target before the compute loop
    // (gfx1250 global_prefetch_b8; all 2*Bcnt addresses are known up front).
    for (int b = 0; b < Bcnt; ++b) {
        __builtin_prefetch(x + (long long)b * N3 + so,  0, 3);
        __builtin_prefetch(x + (long long)b * N3 + dof, 0, 3);
    }

#pragma unroll 4
    for (int b = 0; b < Bcnt; ++b) {
        const long long base = (long long)b * N3;
        const float ax = x[base + so + 0];
        const float ay = x[base + so + 1];
        const float az = x[base + so + 2];
        const float bx = x[base + dof + 0];
        const float by = x[base + dof + 1];
        const float bz = x[base + dof + 2];

        const float dx = ax - bx;
        const float dy = ay - by;
        const float dz = az - bz;

        const float dist = sqrtf(dx * dx + dy * dy + dz * dz);
        const float C    = dist - L0;
        // correction = clip( (-C/(denom+eps)) * diff/(dist+eps), +-MAX_CORR )
        const float scale = (C * neg_inv_denom) / (dist + 1e-9f);

        const float cx = fminf(fmaxf(scale * dx, -MAX_CORR), MAX_CORR);
        const float cy = fminf(fmaxf(scale * dy, -MAX_CORR), MAX_CORR);
        const float cz = fminf(fmaxf(scale * dz, -MAX_CORR), MAX_CORR);

        // dx_i = +wi*corr at src ; dx_j = -wj*corr at dst
        atomicAdd(&delta[base + so + 0],  wi * cx);
        atomicAdd(&delta[base + so + 1],  wi * cy);
        atomicAdd(&delta[base + so + 2],  wi * cz);
        atomicAdd(&delta[base + dof + 0], -wj * cx);
        atomicAdd(&delta[base + dof + 1], -wj * cy);
        atomicAdd(&delta[base + dof + 2], -wj * cz);
    }
}

// ---------------------------------------------------------------------------
// apply: x += delta ; delta = 0  (ready for next Jacobi iteration)
// ---------------------------------------------------------------------------
__global__ void xpbd_apply_kernel(float4* __restrict__ x,
                                  float4* __restrict__ delta,
                                  int n4) {
    int i = blockIdx.x * blockDim.x + threadIdx.x;
    if (i < n4) {
        float4 v  = x[i];
        float4 dd = delta[i];
        v.x += dd.x; v.y += dd.y; v.z += dd.z; v.w += dd.w;
        x[i]     = v;
        delta[i] = make_float4(0.f, 0.f, 0.f, 0.f);
    }
}

extern "C" void kernel_launch(void* const* d_in, const int* in_sizes, int n_in,
                              void* d_out, int out_size, void* d_ws, size_t ws_size,
                              hipStream_t stream) {
    // inputs per setup_inputs(): x (B,N,3) f32 ; edge_index (2,E) i32 ;
    // rest_lengths (E,) f32 ; inv_mass (N,) f32
    const float* x_in = (const float*)d_in[0];
    const int*   eidx = (const int*)d_in[1];
    const float* L0   = (const float*)d_in[2];
    const float* invm = (const float*)d_in[3];

    const int E     = in_sizes[2];
    const int N     = in_sizes[3];
    const int total = in_sizes[0];          // B*N*3
    const int Bcnt  = total / (3 * N);      // 16

    float* xout  = (float*)d_out;
    float* delta = (float*)d_ws;            // needs total*4 bytes (19.2 MB)

    const int n4 = total / 4;               // total % 4 == 0 for (16,100000,3)
    const int TB = 256;                     // 8 wave32 per block

    const int gridV = (n4 + TB - 1) / TB;
    const int gridE = (E + EDGE_TILE - 1) / EDGE_TILE;

    xpbd_init_kernel<<<gridV, TB, 0, stream>>>(
        (const float4*)x_in, (float4*)xout, (float4*)delta, n4);

    for (int it = 0; it < N_ITERS; ++it) {
        xpbd_edge_kernel<<<gridE, EDGE_TILE, 0, stream>>>(
            xout, eidx, L0, invm, delta, N, E, Bcnt);
        xpbd_apply_kernel<<<gridV, TB, 0, stream>>>(
            (float4*)xout, (float4*)delta, n4);
    }
}